// FNOLayer_56710748177060
// MI455X (gfx1250) — compile-verified
//
#include <hip/hip_runtime.h>
#include <math.h>

typedef __attribute__((ext_vector_type(2))) float v2f;
typedef __attribute__((ext_vector_type(8))) float v8f;

#define TWO_PI_256 (6.283185307179586f / 256.f)

__device__ __forceinline__ v8f wmma4(v2f a, v2f b, v8f c) {
  return __builtin_amdgcn_wmma_f32_16x16x4_f32(false, a, false, b, (short)0, c, false, false);
}

// ---------------------------------------------------------------------------
// K0: zero the GroupNorm accumulators
// ---------------------------------------------------------------------------
__global__ void k0_zero(float* __restrict__ p, int n) {
  int i = blockIdx.x * 256 + threadIdx.x;
  if (i < n) p[i] = 0.f;
}

// ---------------------------------------------------------------------------
// K1: partial DFT along W.  X1[row][2*ky+{re,im}] = sum_w x[row][w] e^{-2pi i w ky/256}
// rows = (b*64+c)*256+h, 262144 rows.  GEMM M=16 rows, K=256(w), N=32 via WMMA.
// ---------------------------------------------------------------------------
__global__ __launch_bounds__(128) void k1_dftw(const float* __restrict__ x,
                                               float* __restrict__ X1) {
  __shared__ float xs[64 * 256];   // 64 rows of x
  __shared__ float E[256 * 32];    // twiddles [w][n]
  int tid = threadIdx.x;
  for (int idx = tid; idx < 256 * 32; idx += 128) {
    int w = idx >> 5, n = idx & 31, ky = n >> 1;
    float s, c;
    __sincosf((float)((w * ky) & 255) * TWO_PI_256, &s, &c);
    E[idx] = (n & 1) ? -s : c;
  }
  size_t rbase = (size_t)blockIdx.x * 64;
  for (int idx = tid; idx < 64 * 256; idx += 128) xs[idx] = x[rbase * 256 + idx];
  __syncthreads();

  int wave = tid >> 5, lane = tid & 31, lm = lane & 15, kg = lane >> 4;
  int rl = wave * 16;
  v8f acc0 = {}, acc1 = {};
  for (int k0 = 0; k0 < 256; k0 += 4) {
    int kA = k0 + kg * 2;
    v2f a;  a.x  = xs[(rl + lm) * 256 + kA];     a.y  = xs[(rl + lm) * 256 + kA + 1];
    v2f b0; b0.x = E[kA * 32 + lm];              b0.y = E[(kA + 1) * 32 + lm];
    v2f b1; b1.x = E[kA * 32 + 16 + lm];         b1.y = E[(kA + 1) * 32 + 16 + lm];
    acc0 = wmma4(a, b0, acc0);
    acc1 = wmma4(a, b1, acc1);
  }
  size_t row0 = rbase + rl + kg * 8;
#pragma unroll
  for (int v = 0; v < 8; v++) {
    X1[(row0 + v) * 32 + lm]      = acc0[v];
    X1[(row0 + v) * 32 + 16 + lm] = acc1[v];
  }
}

// ---------------------------------------------------------------------------
// K2: complex DFT along H for the 32 kept kx modes (kx 0..15 and 240..255).
// One block per (b,i).  X2 planes laid out [mode][i][b].
// ---------------------------------------------------------------------------
__global__ __launch_bounds__(256) void k2_dfth(const float* __restrict__ X1,
                                               float* __restrict__ X2re,
                                               float* __restrict__ X2im) {
  __shared__ float xr[256 * 16], xi[256 * 16];
  __shared__ float ct[256], st[256];
  int tid = threadIdx.x;
  int bi = blockIdx.x, b = bi >> 6, ci = bi & 63;
  {
    float s, c;
    __sincosf((float)tid * TWO_PI_256, &s, &c);
    ct[tid] = c; st[tid] = s;
  }
  for (int idx = tid; idx < 256 * 32; idx += 256) {
    int h = idx >> 5, n = idx & 31;
    float v = X1[((size_t)bi * 256 + h) * 32 + n];
    if (n & 1) xi[h * 16 + (n >> 1)] = v; else xr[h * 16 + (n >> 1)] = v;
  }
  __syncthreads();
  for (int t = tid; t < 512; t += 256) {
    int kx = t >> 4, ky = t & 15;
    int kxa = (kx < 16) ? kx : (224 + kx);   // 240 + (kx-16)
    float sre = 0.f, sim = 0.f;
    for (int h = 0; h < 256; h++) {
      int p = (h * kxa) & 255;
      float c = ct[p], s = st[p];
      float re = xr[h * 16 + ky], im = xi[h * 16 + ky];
      sre += re * c + im * s;   // * e^{-i theta}
      sim += im * c - re * s;
    }
    X2re[(size_t)t * 1024 + ci * 16 + b] = sre;
    X2im[(size_t)t * 1024 + ci * 16 + b] = sim;
  }
}

// ---------------------------------------------------------------------------
// K3: per-mode complex channel mix: out[b,o] = sum_i X2[b,i] * w[i,o].
// One block (4 waves) per mode; WMMA M=16(b), K=64(i), N=64(o).
// F32 WMMA has no A/B-negate modifier, so negate the Aim fragment in VALU.
// ---------------------------------------------------------------------------
__global__ __launch_bounds__(128) void k3_mix(const float* __restrict__ X2re,
                                              const float* __restrict__ X2im,
                                              const float* __restrict__ w1re,
                                              const float* __restrict__ w1im,
                                              const float* __restrict__ w2re,
                                              const float* __restrict__ w2im,
                                              float* __restrict__ Ore,
                                              float* __restrict__ Oim) {
  __shared__ float Ar[64 * 16], Ai[64 * 16];   // [i][b]
  __shared__ float Br[64 * 64], Bi[64 * 64];   // [i][o]
  int tid = threadIdx.x;
  int mode = blockIdx.x, kx = mode >> 4, ky = mode & 15;
  const float* wre; const float* wim; int wm;
  if (kx < 16) { wre = w1re; wim = w1im; wm = mode; }
  else         { wre = w2re; wim = w2im; wm = (kx - 16) * 16 + ky; }
  for (int idx = tid; idx < 1024; idx += 128) {
    Ar[idx] = X2re[(size_t)mode * 1024 + idx];
    Ai[idx] = X2im[(size_t)mode * 1024 + idx];
  }
  for (int idx = tid; idx < 4096; idx += 128) {   // idx = i*64+o
    Br[idx] = wre[(size_t)idx * 256 + wm];
    Bi[idx] = wim[(size_t)idx * 256 + wm];
  }
  __syncthreads();
  int wave = tid >> 5, lane = tid & 31, lm = lane & 15, kg = lane >> 4;
  int nt = wave;                       // o-tile
  v8f accRe = {}, accIm = {};
  for (int k0 = 0; k0 < 64; k0 += 4) {
    int kA = k0 + kg * 2;
    v2f are; are.x = Ar[kA * 16 + lm];        are.y = Ar[(kA + 1) * 16 + lm];
    v2f aim; aim.x = Ai[kA * 16 + lm];        aim.y = Ai[(kA + 1) * 16 + lm];
    v2f ain; ain.x = -aim.x;                  ain.y = -aim.y;
    v2f bre; bre.x = Br[kA * 64 + nt * 16 + lm]; bre.y = Br[(kA + 1) * 64 + nt * 16 + lm];
    v2f bim; bim.x = Bi[kA * 64 + nt * 16 + lm]; bim.y = Bi[(kA + 1) * 64 + nt * 16 + lm];
    accRe = wmma4(are, bre, accRe);
    accRe = wmma4(ain, bim, accRe);   // - Aim*Bim
    accIm = wmma4(are, bim, accIm);
    accIm = wmma4(aim, bre, accIm);
  }
#pragma unroll
  for (int v = 0; v < 8; v++) {        // D: b = v+8*kg, o = nt*16+lm
    Ore[(size_t)mode * 1024 + (nt * 16 + lm) * 16 + v + 8 * kg] = accRe[v];
    Oim[(size_t)mode * 1024 + (nt * 16 + lm) * 16 + v + 8 * kg] = accIm[v];
  }
}

// ---------------------------------------------------------------------------
// K4: inverse DFT along H (32 modes -> 256 h), applies 1/H.
// One block per (b,o); Y1[row][2*ky+{re,im}], row=(b*64+o)*256+h.
// ---------------------------------------------------------------------------
__global__ __launch_bounds__(256) void k4_idfth(const float* __restrict__ Ore,
                                                const float* __restrict__ Oim,
                                                float* __restrict__ Y1) {
  __shared__ float orr[512], oii[512];
  __shared__ float ct[256], st[256];
  int tid = threadIdx.x;
  int bi = blockIdx.x, b = bi >> 6, o = bi & 63;
  {
    float s, c;
    __sincosf((float)tid * TWO_PI_256, &s, &c);
    ct[tid] = c; st[tid] = s;
  }
  for (int m = tid; m < 512; m += 256) {
    orr[m] = Ore[(size_t)m * 1024 + o * 16 + b];
    oii[m] = Oim[(size_t)m * 1024 + o * 16 + b];
  }
  __syncthreads();
  int h = tid;
  float out[32];
#pragma unroll
  for (int n = 0; n < 32; n++) out[n] = 0.f;
  for (int kx = 0; kx < 32; kx++) {
    int kxa = (kx < 16) ? kx : (224 + kx);
    int p = (h * kxa) & 255;
    float c = ct[p], s = st[p];
#pragma unroll
    for (int ky = 0; ky < 16; ky++) {
      float re = orr[kx * 16 + ky], im = oii[kx * 16 + ky];
      out[2 * ky]     += re * c - im * s;   // * e^{+i theta}
      out[2 * ky + 1] += im * c + re * s;
    }
  }
  size_t row = (size_t)bi * 256 + h;
  const float sc = 1.f / 256.f;
#pragma unroll
  for (int n = 0; n < 32; n++) Y1[row * 32 + n] = out[n] * sc;
}

// ---------------------------------------------------------------------------
// K5: fused irfft-along-W + 1x1 conv + bias + residual + exact GELU,
// plus per-(b,group) sum/sumsq accumulation.  One block per (b,h).
// Both GEMMs accumulate into the same WMMA accumulator.
// ---------------------------------------------------------------------------
__global__ __launch_bounds__(256) void k5_fused(const float* __restrict__ x,
                                                const float* __restrict__ Y1,
                                                const float* __restrict__ convw,
                                                const float* __restrict__ convb,
                                                float* __restrict__ out,
                                                float* __restrict__ gsum,
                                                float* __restrict__ gsq) {
  __shared__ float xs[64 * 256];   // [i][w]
  __shared__ float T[32 * 256];    // irfft twiddles [k][w]
  __shared__ float Wc[64 * 64];    // conv_w [o][i]
  __shared__ float Ya[64 * 32];    // Y1 rows [o][k]
  __shared__ float ls[32], lq[32];
  int tid = threadIdx.x;
  int bh = blockIdx.x, b = bh >> 8, h = bh & 255;

  for (int idx = tid; idx < 32 * 256; idx += 256) {
    int k = idx >> 8, w = idx & 255, ky = k >> 1;
    float s, c;
    __sincosf((float)((w * ky) & 255) * TWO_PI_256, &s, &c);
    float sc = (ky == 0 ? 1.f : 2.f) * (1.f / 256.f);
    float val = (k & 1) ? -s * sc : c * sc;
    if (k == 1) val = 0.f;   // irfft drops Im of bin 0
    T[idx] = val;
  }
  for (int idx = tid; idx < 64 * 256; idx += 256)
    xs[idx] = x[(((size_t)b * 64 + (idx >> 8)) * 256 + h) * 256 + (idx & 255)];
  for (int idx = tid; idx < 4096; idx += 256) Wc[idx] = convw[idx];
  for (int idx = tid; idx < 2048; idx += 256)
    Ya[idx] = Y1[(((size_t)b * 64 + (idx >> 5)) * 256 + h) * 32 + (idx & 31)];
  if (tid < 32) { ls[tid] = 0.f; lq[tid] = 0.f; }
  __syncthreads();

  int wave = tid >> 5, lane = tid & 31, lm = lane & 15, kg = lane >> 4;
  for (int t = wave; t < 64; t += 8) {
    int ot = t >> 4, wt = t & 15;
    v8f acc = {};
    for (int k0 = 0; k0 < 32; k0 += 4) {          // spectral GEMM, K=32
      int kA = k0 + kg * 2;
      v2f a;  a.x  = Ya[(ot * 16 + lm) * 32 + kA]; a.y = Ya[(ot * 16 + lm) * 32 + kA + 1];
      v2f bb; bb.x = T[kA * 256 + wt * 16 + lm];   bb.y = T[(kA + 1) * 256 + wt * 16 + lm];
      acc = wmma4(a, bb, acc);
    }
    for (int k0 = 0; k0 < 64; k0 += 4) {          // conv GEMM, K=64
      int kA = k0 + kg * 2;
      v2f a;  a.x  = Wc[(ot * 16 + lm) * 64 + kA]; a.y = Wc[(ot * 16 + lm) * 64 + kA + 1];
      v2f bb; bb.x = xs[kA * 256 + wt * 16 + lm];  bb.y = xs[(kA + 1) * 256 + wt * 16 + lm];
      acc = wmma4(a, bb, acc);
    }
    int w = wt * 16 + lm;
#pragma unroll
    for (int v = 0; v < 8; v++) {
      int o = ot * 16 + v + 8 * kg;
      float val = acc[v] + convb[o] + xs[o * 256 + w];
      float g = 0.5f * val * (1.f + erff(val * 0.70710678118654752f));
      out[(((size_t)b * 64 + o) * 256 + h) * 256 + w] = g;
      atomicAdd(&ls[o >> 1], g);
      atomicAdd(&lq[o >> 1], g * g);
    }
  }
  __syncthreads();
  if (tid < 32) {
    atomicAdd(&gsum[b * 32 + tid], ls[tid]);
    atomicAdd(&gsq [b * 32 + tid], lq[tid]);
  }
}

// ---------------------------------------------------------------------------
// K6: GroupNorm finalize, in place over d_out.
// ---------------------------------------------------------------------------
__global__ __launch_bounds__(256) void k6_gn(float* __restrict__ out,
                                             const float* __restrict__ gsum,
                                             const float* __restrict__ gsq,
                                             const float* __restrict__ gw,
                                             const float* __restrict__ gb) {
  size_t idx = (size_t)blockIdx.x * 256 + threadIdx.x;
  int c = (int)((idx >> 16) & 63);
  int b = (int)(idx >> 22);
  int g = c >> 1;
  const float n = 131072.f;   // 2 * 256 * 256
  float mu  = gsum[b * 32 + g] / n;
  float var = gsq [b * 32 + g] / n - mu * mu;
  float inv = rsqrtf(var + 1e-5f);
  float v = out[idx];
  out[idx] = (v - mu) * inv * gw[c] + gb[c];
}

// ---------------------------------------------------------------------------
extern "C" void kernel_launch(void* const* d_in, const int* in_sizes, int n_in,
                              void* d_out, int out_size, void* d_ws, size_t ws_size,
                              hipStream_t stream) {
  const float* x     = (const float*)d_in[0];
  const float* w1re  = (const float*)d_in[1];
  const float* w1im  = (const float*)d_in[2];
  const float* w2re  = (const float*)d_in[3];
  const float* w2im  = (const float*)d_in[4];
  const float* convw = (const float*)d_in[5];
  const float* convb = (const float*)d_in[6];
  const float* gw    = (const float*)d_in[7];
  const float* gb    = (const float*)d_in[8];
  float* out = (float*)d_out;
  float* ws  = (float*)d_ws;

  float* X1   = ws;                    // 8388608 floats
  float* Y1   = ws + 8388608;          // 8388608
  float* X2re = ws + 16777216;         // 524288
  float* X2im = X2re + 524288;
  float* Ore  = X2im + 524288;
  float* Oim  = Ore  + 524288;
  float* gsum = Oim  + 524288;         // 512
  float* gsq  = gsum + 512;            // 512

  k0_zero <<<4,      256, 0, stream>>>(gsum, 1024);
  k1_dftw <<<4096,   128, 0, stream>>>(x, X1);
  k2_dfth <<<1024,   256, 0, stream>>>(X1, X2re, X2im);
  k3_mix  <<<512,    128, 0, stream>>>(X2re, X2im, w1re, w1im, w2re, w2im, Ore, Oim);
  k4_idfth<<<1024,   256, 0, stream>>>(Ore, Oim, Y1);
  k5_fused<<<4096,   256, 0, stream>>>(x, Y1, convw, convb, out, gsum, gsq);
  k6_gn   <<<262144, 256, 0, stream>>>(out, gsum, gsq, gw, gb);
}